// BNNHAN_41841571397955
// MI455X (gfx1250) — compile-verified
//
#include <hip/hip_runtime.h>
#include <hip/hip_bf16.h>

typedef __attribute__((ext_vector_type(2))) float v2f;
typedef __attribute__((ext_vector_type(8))) float v8f;

// ---------------- utility ----------------

__global__ void fill_kernel(unsigned* __restrict__ p, unsigned v, size_t cnt) {
    size_t i = (size_t)blockIdx.x * blockDim.x + threadIdx.x;
    size_t stride = (size_t)gridDim.x * blockDim.x;
    for (; i < cnt; i += stride) p[i] = v;
}

__global__ void relu_kernel(float* __restrict__ p, size_t cnt) {
    size_t i = (size_t)blockIdx.x * blockDim.x + threadIdx.x;
    size_t stride = (size_t)gridDim.x * blockDim.x;
    for (; i < cnt; i += stride) p[i] = fmaxf(p[i], 0.0f);
}

// ordered-uint mapping so unsigned atomicMax == float max (handles negatives)
__device__ __forceinline__ unsigned fmap(float f) {
    unsigned u = __float_as_uint(f);
    return (u & 0x80000000u) ? ~u : (u | 0x80000000u);
}
__device__ __forceinline__ float funmap(unsigned u) {
    return __uint_as_float((u & 0x80000000u) ? (u & 0x7fffffffu) : ~u);
}
__device__ __forceinline__ float leaky(float v) { return v >= 0.0f ? v : 0.2f * v; }

// ---------------- 1) projection GEMM: h = x @ proj_w^T + b  (fp32 WMMA) ----------------
// grid.x = n/16 (n == 100000 -> exactly 6250), block = 256 (8 waves).
// Each block: one 16-row M tile staged in LDS; wave w computes output cols [16w,16w+16).
// K = 128 consumed in 32 steps of V_WMMA_F32_16X16X4_F32. EXEC is all-ones throughout.
__global__ __launch_bounds__(256)
void proj_wmma_kernel(const float* __restrict__ x, const float* __restrict__ w,
                      const float* __restrict__ b, float* __restrict__ h) {
    __shared__ float As[16 * 132];                       // padded stride kills bank conflicts
    const int m0 = blockIdx.x * 16;
    const int tid = threadIdx.x;

    // cooperative load of 16x128 x-tile (2 float4 per thread)
    int q = tid * 2;
#pragma unroll
    for (int it = 0; it < 2; ++it, ++q) {
        int row = q >> 5;                                // 32 float4 per 128-float row
        int c4 = (q & 31) << 2;
        float4 v = *(const float4*)(x + (size_t)(m0 + row) * 128 + c4);
        As[row * 132 + c4 + 0] = v.x;
        As[row * 132 + c4 + 1] = v.y;
        As[row * 132 + c4 + 2] = v.z;
        As[row * 132 + c4 + 3] = v.w;
    }
    __syncthreads();

    const int wv = tid >> 5;                             // wave id -> 16-col tile
    const int lane = tid & 31;
    const int mrow = lane & 15;
    const int khalf = (lane >> 4) << 1;                  // 0 or 2 (K pair select)
    const int coln = wv * 16 + (lane & 15);
    const float* wrow = w + (size_t)coln * 128;          // proj_w[coln][k] == B[k][coln]

    v8f acc = {};
#pragma unroll 4
    for (int kk = 0; kk < 128; kk += 4) {
        const int kb = kk + khalf;
        v2f a, bm;
        a.x = As[mrow * 132 + kb];
        a.y = As[mrow * 132 + kb + 1];
        bm.x = wrow[kb];
        bm.y = wrow[kb + 1];
        acc = __builtin_amdgcn_wmma_f32_16x16x4_f32(false, a, false, bm,
                                                    (short)0, acc, false, false);
    }
    const float bias = b[coln];
#pragma unroll
    for (int v = 0; v < 8; ++v) {
        int mr = v + ((lane >> 4) << 3);                 // C/D layout: M = v + 8*(lane/16)
        h[(size_t)(m0 + mr) * 128 + coln] = acc[v] + bias;
    }
}

// ---------------- 2) per-node attention logits a_src/a_dst [R,N,H] ----------------
__global__ void att_kernel(const float* __restrict__ h, const float* __restrict__ att_src,
                           const float* __restrict__ att_dst, float* __restrict__ aS,
                           float* __restrict__ aD, int n) {
    int idx = blockIdx.x * blockDim.x + threadIdx.x;
    if (idx >= n * 8) return;
    int node = idx >> 3, hd = idx & 7;
    float hv[16];
    const float* hp = h + (size_t)node * 128 + hd * 16;
#pragma unroll
    for (int i = 0; i < 16; i += 4) {
        float4 v = *(const float4*)(hp + i);
        hv[i] = v.x; hv[i + 1] = v.y; hv[i + 2] = v.z; hv[i + 3] = v.w;
    }
#pragma unroll
    for (int r = 0; r < 3; ++r) {
        const float* ws = att_src + (r * 8 + hd) * 16;
        const float* wd = att_dst + (r * 8 + hd) * 16;
        float s = 0.0f, d = 0.0f;
#pragma unroll
        for (int i = 0; i < 16; ++i) { s += hv[i] * ws[i]; d += hv[i] * wd[i]; }
        aS[((size_t)r * n + node) * 8 + hd] = s;
        aD[((size_t)r * n + node) * 8 + hd] = d;
    }
}

// ---------------- 3a) segment max over destinations ----------------
__global__ void edge_max_kernel(const int* __restrict__ ei, const float* __restrict__ aS,
                                const float* __restrict__ aD, unsigned* __restrict__ mbuf,
                                int e) {
    int ed = blockIdx.x * blockDim.x + threadIdx.x;
    if (ed >= e) return;
    int row = ei[ed], col = ei[e + ed];
#pragma unroll
    for (int hd = 0; hd < 8; ++hd) {
        float al = leaky(aS[(size_t)row * 8 + hd] + aD[(size_t)col * 8 + hd]);
        atomicMax(mbuf + (size_t)col * 8 + hd, fmap(al));
    }
}

// ---------------- 3b) segment sum of exp(alpha - m) ----------------
__global__ void edge_sum_kernel(const int* __restrict__ ei, const float* __restrict__ aS,
                                const float* __restrict__ aD, const unsigned* __restrict__ mbuf,
                                float* __restrict__ denom, int e) {
    int ed = blockIdx.x * blockDim.x + threadIdx.x;
    if (ed >= e) return;
    int row = ei[ed], col = ei[e + ed];
#pragma unroll
    for (int hd = 0; hd < 8; ++hd) {
        float al = leaky(aS[(size_t)row * 8 + hd] + aD[(size_t)col * 8 + hd]);
        float m = funmap(mbuf[(size_t)col * 8 + hd]);
        if (!(m > -__builtin_inff() && m < __builtin_inff())) m = 0.0f;   // isfinite -> 0
        atomicAdd(denom + (size_t)col * 8 + hd, expf(al - m));
    }
}

// ---------------- 3c) weighted aggregation, one wave32 per edge ----------------
// lane handles features [4*lane, 4*lane+4) -> head = lane>>2 (all 4 feats same head).
__global__ __launch_bounds__(256)
void edge_agg_kernel(const int* __restrict__ ei, const float* __restrict__ aS,
                     const float* __restrict__ aD, const unsigned* __restrict__ mbuf,
                     const float* __restrict__ denom, const float* __restrict__ h,
                     float* __restrict__ agg, int e) {
    int gt = blockIdx.x * blockDim.x + threadIdx.x;
    int ed = gt >> 5;
    int lane = gt & 31;
    if (ed >= e) return;
    int row = ei[ed], col = ei[e + ed];
    int hd = lane >> 2;
    size_t ci = (size_t)col * 8 + hd;
    float al = leaky(aS[(size_t)row * 8 + hd] + aD[ci]);
    float m = funmap(mbuf[ci]);
    if (!(m > -__builtin_inff() && m < __builtin_inff())) m = 0.0f;
    float wgt = expf(al - m) / (denom[ci] + 1e-16f);
    float4 hv = *(const float4*)(h + (size_t)row * 128 + lane * 4);
    float* ap = agg + (size_t)col * 128 + lane * 4;
    atomicAdd(ap + 0, hv.x * wgt);
    atomicAdd(ap + 1, hv.y * wgt);
    atomicAdd(ap + 2, hv.z * wgt);
    atomicAdd(ap + 3, hv.w * wgt);
}

// ---------------- 4) semantic score: score[r] = (1/N) sum_n q . tanh(outs_r[n] @ k_w^T + k_b)
// One wave per node (grid-stride); lane covers 4 output features; wave-shuffle reduce.
__global__ __launch_bounds__(256)
void semantic_kernel(const float* __restrict__ outs_r, const float* __restrict__ kw,
                     const float* __restrict__ kb, const float* __restrict__ qv,
                     float* __restrict__ score_r, int n, float invn) {
    int gt = blockIdx.x * blockDim.x + threadIdx.x;
    int wid = gt >> 5;
    int lane = gt & 31;
    int nwaves = (gridDim.x * blockDim.x) >> 5;
    float acc = 0.0f;
    for (int node = wid; node < n; node += nwaves) {
        const float* ov = outs_r + (size_t)node * 128;
#pragma unroll
        for (int j = 0; j < 4; ++j) {
            int t = lane * 4 + j;
            const float* kr = kw + (size_t)t * 128;
            float s = kb[t];
            for (int f = 0; f < 128; f += 4) {
                float4 o4 = *(const float4*)(ov + f);
                float4 k4 = *(const float4*)(kr + f);
                s += o4.x * k4.x + o4.y * k4.y + o4.z * k4.z + o4.w * k4.w;
            }
            acc += tanhf(s) * qv[t];
        }
    }
    for (int off = 16; off > 0; off >>= 1) acc += __shfl_down(acc, off, 32);
    if (lane == 0) atomicAdd(score_r, acc * invn);
}

// ---------------- 5) softmax over relations + fuse + final linear ----------------
__global__ void fuse_kernel(const float* __restrict__ outs, const float* __restrict__ score,
                            const float* __restrict__ lin_w, const float* __restrict__ lin_b,
                            float* __restrict__ out, int n) {
    int node = blockIdx.x * blockDim.x + threadIdx.x;
    if (node >= n) return;
    float s0 = score[0], s1 = score[1], s2 = score[2];
    float sm = fmaxf(s0, fmaxf(s1, s2));
    float e0 = expf(s0 - sm), e1 = expf(s1 - sm), e2 = expf(s2 - sm);
    float inv = 1.0f / (e0 + e1 + e2);
    float a0 = e0 * inv, a1 = e1 * inv, a2 = e2 * inv;
    const float* o0 = outs + (size_t)node * 128;
    const float* o1 = o0 + (size_t)n * 128;
    const float* o2 = o1 + (size_t)n * 128;
    float acc[4] = {0.0f, 0.0f, 0.0f, 0.0f};
    for (int f = 0; f < 128; ++f) {
        float fv = a0 * o0[f] + a1 * o1[f] + a2 * o2[f];
#pragma unroll
        for (int o = 0; o < 4; ++o) acc[o] += fv * lin_w[o * 128 + f];
    }
#pragma unroll
    for (int o = 0; o < 4; ++o) out[node * 4 + o] = acc[o] + lin_b[o];
}

// ---------------- launcher ----------------
extern "C" void kernel_launch(void* const* d_in, const int* in_sizes, int n_in,
                              void* d_out, int out_size, void* d_ws, size_t ws_size,
                              hipStream_t stream) {
    const float* x       = (const float*)d_in[0];
    const int*   ei[3]   = {(const int*)d_in[1], (const int*)d_in[2], (const int*)d_in[3]};
    const float* proj_w  = (const float*)d_in[4];
    const float* proj_b  = (const float*)d_in[5];
    const float* att_src = (const float*)d_in[6];
    const float* att_dst = (const float*)d_in[7];
    const float* k_w     = (const float*)d_in[8];
    const float* k_b     = (const float*)d_in[9];
    const float* qv      = (const float*)d_in[10];
    const float* lin_w   = (const float*)d_in[11];
    const float* lin_b   = (const float*)d_in[12];
    float* out = (float*)d_out;

    const int n = in_sizes[0] / 128;    // 100000 (divisible by 16)
    const int e = in_sizes[1] / 2;      // 1000000

    // workspace layout (floats)
    float* ws = (float*)d_ws;
    size_t off = 0;
    float* h    = ws + off; off += (size_t)n * 128;        // [N,128]
    float* outs = ws + off; off += (size_t)3 * n * 128;    // [3,N,128] (agg -> relu in place)
    float* aS   = ws + off; off += (size_t)3 * n * 8;      // [3,N,8]
    float* aD   = ws + off; off += (size_t)3 * n * 8;
    unsigned* mbuf = (unsigned*)(ws + off); off += (size_t)3 * n * 8;  // ordered-uint max
    float* denom = ws + off; off += (size_t)3 * n * 8;
    float* score = ws + off; off += 4;

    // init: agg=0, denom=0 (+score), m = mapped(-inf) = 0x007FFFFF
    fill_kernel<<<2048, 256, 0, stream>>>((unsigned*)outs, 0u, (size_t)3 * n * 128);
    fill_kernel<<<512, 256, 0, stream>>>(mbuf, 0x007FFFFFu, (size_t)3 * n * 8);
    fill_kernel<<<512, 256, 0, stream>>>((unsigned*)denom, 0u, (size_t)3 * n * 8 + 4);

    // projection (fp32 WMMA) + per-node attention logits
    proj_wmma_kernel<<<n / 16, 256, 0, stream>>>(x, proj_w, proj_b, h);
    att_kernel<<<(n * 8 + 255) / 256, 256, 0, stream>>>(h, att_src, att_dst, aS, aD, n);

    // per-relation edge softmax + aggregation
    const int eb = (e + 255) / 256;
    for (int r = 0; r < 3; ++r) {
        const float* aSr = aS + (size_t)r * n * 8;
        const float* aDr = aD + (size_t)r * n * 8;
        unsigned* mr = mbuf + (size_t)r * n * 8;
        float* dr = denom + (size_t)r * n * 8;
        float* aggr = outs + (size_t)r * n * 128;
        edge_max_kernel<<<eb, 256, 0, stream>>>(ei[r], aSr, aDr, mr, e);
        edge_sum_kernel<<<eb, 256, 0, stream>>>(ei[r], aSr, aDr, mr, dr, e);
        edge_agg_kernel<<<(e * 32 + 255) / 256, 256, 0, stream>>>(ei[r], aSr, aDr, mr, dr,
                                                                  h, aggr, e);
    }
    relu_kernel<<<4096, 256, 0, stream>>>(outs, (size_t)3 * n * 128);

    // semantic attention scores, then fuse + final linear
    for (int r = 0; r < 3; ++r)
        semantic_kernel<<<1024, 256, 0, stream>>>(outs + (size_t)r * n * 128, k_w, k_b, qv,
                                                  score + r, n, 1.0f / (float)n);
    fuse_kernel<<<(n + 255) / 256, 256, 0, stream>>>(outs, score, lin_w, lin_b, out, n);
}